// SimpleGCN_76012331205242
// MI455X (gfx1250) — compile-verified
//
#include <hip/hip_runtime.h>
#include <hip/hip_bf16.h>
#include <math.h>

// ---------------------------------------------------------------------------
// SimpleGCN forward for MI455X (gfx1250, wave32, WMMA).
//   3x [ H = X@W (WMMA f32 16x16x4) ; agg = scatter(norm * H[src]) ;
//        X' = elu(agg + dinv^2*H + b) ]
//   then segment-mean pool over batch and pooled @ Wl + bl (WMMA, fused bias).
// Scatter-add uses native global_atomic_add_f32 (no-return, device scope)
// forced via inline asm so the L2-resident accumulation is single-op per
// element instead of a CAS loop.
// ---------------------------------------------------------------------------

#define N_NODES  50000
#define N_EDGES  800000
#define N_GRAPHS 64
#define FDIM     128

typedef __attribute__((ext_vector_type(2))) float v2f;
typedef __attribute__((ext_vector_type(8))) float v8f;

// Native f32 atomic add, no return value (STOREcnt-tracked), device scope so
// the RMW is performed at the device-coherent point (global L2).
__device__ __forceinline__ void atomic_add_f32(float* p, float v) {
    asm volatile("global_atomic_add_f32 %0, %1, off scope:SCOPE_DEV"
                 :
                 : "v"(p), "v"(v)
                 : "memory");
}

// ---------------- zero fill ----------------
__global__ void gcn_fill0_kernel(float* __restrict__ p, int n) {
    int t = blockIdx.x * blockDim.x + threadIdx.x;
    if (t < n) p[t] = 0.0f;
}

// ---------------- degree count (real edges; self-loop added analytically) ---
__global__ void gcn_deg_kernel(const int* __restrict__ dst, float* __restrict__ deg) {
    int e = blockIdx.x * blockDim.x + threadIdx.x;
    if (e < N_EDGES) atomic_add_f32(&deg[dst[e]], 1.0f);
}

// deg -> dinv = rsqrt(deg + 1)   (self-loop guarantees deg >= 1)
__global__ void gcn_dinv_kernel(float* __restrict__ d) {
    int i = blockIdx.x * blockDim.x + threadIdx.x;
    if (i < N_NODES) d[i] = rsqrtf(d[i] + 1.0f);
}

// ---------------- WMMA GEMM: C[rows x 128] = A[rows x 128] @ W[128 x 128] ---
// One wave computes a 16x16 tile; block = 256 threads = 8 waves = one 16-row
// stripe across all 128 columns. rows must be a multiple of 16 (50000, 64 ok).
template <bool ADD_BIAS>
__global__ void gcn_gemm128_wmma_kernel(const float* __restrict__ A,
                                        const float* __restrict__ W,
                                        const float* __restrict__ bias,
                                        float* __restrict__ C) {
    const int lane = threadIdx.x & 31;
    const int wave = threadIdx.x >> 5;
    const int half = lane >> 4;   // 0: K pair {0,1}; 1: K pair {2,3}
    const int lr   = lane & 15;
    const int rowBase = blockIdx.x << 4;
    const int colBase = wave << 4;

    // A fragment source: row (rowBase+lr), contiguous float2 at col k+2*half
    const float* arow = A + (size_t)(rowBase + lr) * FDIM + 2 * half;
    // B fragment source: W[(k+2*half)][colBase+lr]
    const float* wcol = W + (size_t)(2 * half) * FDIM + colBase + lr;

    v8f acc = {};
#pragma unroll
    for (int k = 0; k < FDIM; k += 4) {
        v2f a = *(const v2f*)(arow + k);
        v2f b;
        b.x = wcol[(size_t)k * FDIM];
        b.y = wcol[(size_t)k * FDIM + FDIM];
        // (neg_a, A, neg_b, B, c_mod, C, reuse_a, reuse_b)
        acc = __builtin_amdgcn_wmma_f32_16x16x4_f32(
            false, a, false, b, (short)0, acc, false, false);
    }

    const float bv = ADD_BIAS ? bias[colBase + lr] : 0.0f;
    float* crow = C + (size_t)(rowBase + 8 * half) * FDIM + colBase + lr;
#pragma unroll
    for (int r = 0; r < 8; ++r)
        crow[(size_t)r * FDIM] = acc[r] + bv;
}

// ---------------- edge scatter: agg[dst] += dinv[s]*dinv[d] * h[src] --------
// One wave per edge; lane handles a float4 (4 columns). 128 native f32
// atomics per edge into the 25.6 MB (L2-resident) destination buffer.
__global__ void gcn_scatter_kernel(const float* __restrict__ h,
                                   const int* __restrict__ src,
                                   const int* __restrict__ dst,
                                   const float* __restrict__ dinv,
                                   float* __restrict__ agg) {
    int e = blockIdx.x * 8 + (threadIdx.x >> 5);
    if (e >= N_EDGES) return;
    const int lane = threadIdx.x & 31;
    const int s = src[e];
    const int d = dst[e];
    const float norm = dinv[s] * dinv[d];
    const float4 v = *((const float4*)(h + (size_t)s * FDIM) + lane);
    float* base = agg + (size_t)d * FDIM + lane * 4;
    atomic_add_f32(base + 0, v.x * norm);
    atomic_add_f32(base + 1, v.y * norm);
    atomic_add_f32(base + 2, v.z * norm);
    atomic_add_f32(base + 3, v.w * norm);
}

// ---------------- finalize: agg = elu(agg + dinv^2 * h + bias) --------------
__global__ void gcn_finalize_kernel(float* __restrict__ agg,
                                    const float* __restrict__ h,
                                    const float* __restrict__ dinv,
                                    const float* __restrict__ bias) {
    int t = blockIdx.x * blockDim.x + threadIdx.x;
    if (t >= N_NODES * FDIM) return;
    const int i = t >> 7;
    const int c = t & (FDIM - 1);
    const float di = dinv[i];
    const float v = agg[t] + di * di * h[t] + bias[c];
    agg[t] = (v > 0.0f) ? v : expm1f(v);
}

// ---------------- pooling: pooled[batch[i]] += x[i]; cnt[batch[i]] += 1 -----
__global__ void gcn_pool_kernel(const float* __restrict__ x,
                                const int* __restrict__ batch,
                                float* __restrict__ pooled,
                                float* __restrict__ cnt) {
    int t = blockIdx.x * blockDim.x + threadIdx.x;
    int node = t >> 5;
    if (node >= N_NODES) return;
    const int lane = t & 31;
    const int g = batch[node];
    const float4 v = *((const float4*)(x + (size_t)node * FDIM) + lane);
    float* base = pooled + (size_t)g * FDIM + lane * 4;
    atomic_add_f32(base + 0, v.x);
    atomic_add_f32(base + 1, v.y);
    atomic_add_f32(base + 2, v.z);
    atomic_add_f32(base + 3, v.w);
    if (lane == 0) atomic_add_f32(&cnt[g], 1.0f);
}

__global__ void gcn_mean_kernel(float* __restrict__ pooled,
                                const float* __restrict__ cnt) {
    int t = blockIdx.x * blockDim.x + threadIdx.x;
    if (t >= N_GRAPHS * FDIM) return;
    pooled[t] /= fmaxf(cnt[t >> 7], 1.0f);
}

// ---------------------------------------------------------------------------
extern "C" void kernel_launch(void* const* d_in, const int* in_sizes, int n_in,
                              void* d_out, int out_size, void* d_ws, size_t ws_size,
                              hipStream_t stream) {
    const float* x     = (const float*)d_in[0];
    const int*   ei    = (const int*)d_in[1];   // [2, E] row-major
    const int*   batch = (const int*)d_in[2];
    const float* W1 = (const float*)d_in[3];
    const float* b1 = (const float*)d_in[4];
    const float* W2 = (const float*)d_in[5];
    const float* b2 = (const float*)d_in[6];
    const float* W3 = (const float*)d_in[7];
    const float* b3 = (const float*)d_in[8];
    const float* Wl = (const float*)d_in[9];
    const float* bl = (const float*)d_in[10];
    float* out = (float*)d_out;

    const int* src = ei;
    const int* dst = ei + N_EDGES;

    // workspace layout (floats)
    float* ws     = (float*)d_ws;
    float* dinv   = ws;                               // N
    float* hbuf   = dinv + N_NODES;                   // N*128
    float* agg    = hbuf + (size_t)N_NODES * FDIM;    // N*128
    float* pooled = agg + (size_t)N_NODES * FDIM;     // G*128
    float* cnt    = pooled + (size_t)N_GRAPHS * FDIM; // G  (contiguous w/ pooled)

    const int BLK = 256;
    const int gridNodes   = (N_NODES + BLK - 1) / BLK;
    const int gridEdges   = (N_EDGES + BLK - 1) / BLK;
    const int gridElems   = (N_NODES * FDIM + BLK - 1) / BLK;       // 25000
    const int gridGemm    = N_NODES / 16;                           // 3125
    const int gridScatter = (N_EDGES + 7) / 8;                      // 100000
    const int gridPool    = (N_NODES * 32 + BLK - 1) / BLK;         // 6250

    // degrees -> dinv
    gcn_fill0_kernel<<<gridNodes, BLK, 0, stream>>>(dinv, N_NODES);
    gcn_deg_kernel<<<gridEdges, BLK, 0, stream>>>(dst, dinv);
    gcn_dinv_kernel<<<gridNodes, BLK, 0, stream>>>(dinv);

    // layer 1
    gcn_gemm128_wmma_kernel<false><<<gridGemm, BLK, 0, stream>>>(x, W1, nullptr, hbuf);
    gcn_fill0_kernel<<<gridElems, BLK, 0, stream>>>(agg, N_NODES * FDIM);
    gcn_scatter_kernel<<<gridScatter, BLK, 0, stream>>>(hbuf, src, dst, dinv, agg);
    gcn_finalize_kernel<<<gridElems, BLK, 0, stream>>>(agg, hbuf, dinv, b1);

    // layer 2
    gcn_gemm128_wmma_kernel<false><<<gridGemm, BLK, 0, stream>>>(agg, W2, nullptr, hbuf);
    gcn_fill0_kernel<<<gridElems, BLK, 0, stream>>>(agg, N_NODES * FDIM);
    gcn_scatter_kernel<<<gridScatter, BLK, 0, stream>>>(hbuf, src, dst, dinv, agg);
    gcn_finalize_kernel<<<gridElems, BLK, 0, stream>>>(agg, hbuf, dinv, b2);

    // layer 3
    gcn_gemm128_wmma_kernel<false><<<gridGemm, BLK, 0, stream>>>(agg, W3, nullptr, hbuf);
    gcn_fill0_kernel<<<gridElems, BLK, 0, stream>>>(agg, N_NODES * FDIM);
    gcn_scatter_kernel<<<gridScatter, BLK, 0, stream>>>(hbuf, src, dst, dinv, agg);
    gcn_finalize_kernel<<<gridElems, BLK, 0, stream>>>(agg, hbuf, dinv, b3);

    // global mean pool
    gcn_fill0_kernel<<<(N_GRAPHS * FDIM + N_GRAPHS + BLK - 1) / BLK, BLK, 0, stream>>>(
        pooled, N_GRAPHS * FDIM + N_GRAPHS);
    gcn_pool_kernel<<<gridPool, BLK, 0, stream>>>(agg, batch, pooled, cnt);
    gcn_mean_kernel<<<(N_GRAPHS * FDIM + BLK - 1) / BLK, BLK, 0, stream>>>(pooled, cnt);

    // classifier: out = pooled @ Wl + bl   (64 rows -> 4 blocks)
    gcn_gemm128_wmma_kernel<true><<<N_GRAPHS / 16, BLK, 0, stream>>>(pooled, Wl, bl, out);
}